// Raymarcher_4612794876580
// MI455X (gfx1250) — compile-verified
//
#include <hip/hip_runtime.h>
#include <stdint.h>

#define HH 128
#define WW 128
#define KK 64
#define MD 16
#define NSTEP 48
#define TW 16
#define THY 8
#define BLOCK (TW * THY)                 // 128 threads = 4 waves (wave32)
#define TPL_FLOATS (4 * MD * MD * MD)    // 16384 floats = 64 KB per prim

#if __has_builtin(__builtin_amdgcn_tensor_load_to_lds)
#define USE_TDM 1
typedef unsigned int v4u __attribute__((ext_vector_type(4)));
typedef int v8i __attribute__((ext_vector_type(8)));
typedef int v4i __attribute__((ext_vector_type(4)));
#else
#define USE_TDM 0
#endif

// ---- CDNA5 async global->LDS copy (ASYNCcnt path) fallback.
__device__ __forceinline__ void async_cp_b128(uint32_t lds_off, const float* gsrc) {
    uint64_t ga = (uint64_t)(uintptr_t)gsrc;
    asm volatile("global_load_async_to_lds_b128 %0, %1, off"
                 :: "v"(lds_off), "v"(ga) : "memory");
}

__device__ __forceinline__ void wait_async0() {
#if __has_builtin(__builtin_amdgcn_s_wait_asynccnt)
    __builtin_amdgcn_s_wait_asynccnt(0);
#else
    asm volatile("s_wait_asynccnt 0" ::: "memory");
#endif
}

__device__ __forceinline__ void wait_tensor0() {
#if __has_builtin(__builtin_amdgcn_s_wait_tensorcnt)
    __builtin_amdgcn_s_wait_tensorcnt(0);
#else
    asm volatile("s_wait_tensorcnt 0" ::: "memory");
#endif
}

// Stage one prim's 64KB template (contiguous) into an LDS buffer.
__device__ __forceinline__ void stage(const float* __restrict__ src,
                                      const float* ldsDst, int tid) {
#if USE_TDM
    // One TDM descriptor: 1-D tile of 16384 x 4B elements. Wave 0 only.
    if ((tid >> 5) == 0) {
        uint64_t ga = (uint64_t)(uintptr_t)src;
        uint32_t lds = (uint32_t)(uintptr_t)ldsDst;       // LDS byte offset
        uint32_t ga_lo = (uint32_t)__builtin_amdgcn_readfirstlane((int)(ga & 0xffffffffu));
        uint32_t ga_hi = (uint32_t)__builtin_amdgcn_readfirstlane((int)(ga >> 32));
        lds = (uint32_t)__builtin_amdgcn_readfirstlane((int)lds);

        v4u g0;
        g0.x = 1u;                                        // count=1, user mode
        g0.y = lds;                                       // lds_addr bits [63:32]
        g0.z = ga_lo;                                     // global_addr bits [95:64]
        g0.w = (ga_hi & 0x01ffffffu) | 0x80000000u;       // addr[56:32] | type=2

        v8i g1;
        g1.s0 = (int)(2u << 16);                          // data_size=4B, wg_mask=0
        g1.s1 = (int)(16384u << 16);                      // tensor_dim0[15:0] @ bits 63:48
        g1.s2 = (int)((16384u >> 16) | (1u << 16));       // tensor_dim0[31:16], tensor_dim1=1
        g1.s3 = (int)(16384u << 16);                      // tile_dim0 = 16384 @ bits 127:112
        g1.s4 = 0;                                        // tile_dim1=0, tile_dim2=0 (unused)
        g1.s5 = 16384;                                    // tensor_dim0_stride[31:0]
        g1.s6 = 0;                                        // stride hi, dim1_stride lo
        g1.s7 = 0;

        v4i g2 = {0, 0, 0, 0};
        v4i g3 = {0, 0, 0, 0};
        v8i g4 = {0, 0, 0, 0, 0, 0, 0, 0};                // clang-23 6-arg form
        __builtin_amdgcn_tensor_load_to_lds(g0, g1, g2, g3, g4, 0);
    }
#else
    // 16384 floats / (128 threads * 4 floats) = 32 b128 async copies per thread.
    uint32_t base = (uint32_t)(uintptr_t)ldsDst;
#pragma unroll
    for (int it = 0; it < TPL_FLOATS / (BLOCK * 4); ++it) {
        int fo = (it * BLOCK + tid) * 4;
        async_cp_b128(base + (uint32_t)fo * 4u, src + fo);
    }
#endif
}

__device__ __forceinline__ void stage_wait(int tid) {
#if USE_TDM
    if ((tid >> 5) == 0) wait_tensor0();    // TENSORcnt is per-wave; wave0 issued
#else
    (void)tid;
    wait_async0();
#endif
}

__device__ __forceinline__ void axclip(float b, float d, float& lo, float& hi) {
    if (fabsf(d) > 1e-12f) {
        float inv = 1.0f / d;
        float t1 = (-1.0f - b) * inv;
        float t2 = ( 1.0f - b) * inv;
        lo = fmaxf(lo, fminf(t1, t2));
        hi = fminf(hi, fmaxf(t1, t2));
    } else if (fabsf(b) > 1.0f) {
        hi = lo - 1.0f;   // empty
    }
}

struct PrimMarch { float gb[3], gd[3]; int slo, shi; };

__device__ __forceinline__ bool prim_setup(
    int k, const float* s_pos, const float* s_rot, const float* s_scl,
    float px, float py, float pz, float dx, float dy, float dz,
    float tmn, float tmx, float dtv, PrimMarch& pm)
{
    const float* R = &s_rot[k * 9];     // R[j*3+i]
    float S0 = s_scl[k * 3 + 0], S1 = s_scl[k * 3 + 1], S2 = s_scl[k * 3 + 2];
    float rx = px - s_pos[k * 3 + 0];
    float ry = py - s_pos[k * 3 + 1];
    float rz = pz - s_pos[k * 3 + 2];
    // local_i = (sum_j R[j][i] * rel_j) * S_i   (einsum 'bkji,bhwkj->bhwki')
    float b0 = (R[0] * rx + R[3] * ry + R[6] * rz) * S0;
    float b1 = (R[1] * rx + R[4] * ry + R[7] * rz) * S1;
    float b2 = (R[2] * rx + R[5] * ry + R[8] * rz) * S2;
    float d0 = (R[0] * dx + R[3] * dy + R[6] * dz) * S0;
    float d1 = (R[1] * dx + R[4] * dy + R[7] * dz) * S1;
    float d2 = (R[2] * dx + R[5] * dy + R[8] * dz) * S2;

    float lo = tmn;
    float hi = fminf(tmx, tmn + (float)(NSTEP - 1) * dtv);
    axclip(b0, d0, lo, hi);
    axclip(b1, d1, lo, hi);
    axclip(b2, d2, lo, hi);
    if (lo > hi) return false;

    int slo = (int)ceilf ((lo - tmn) / dtv - 1e-4f);
    int shi = (int)floorf((hi - tmn) / dtv + 1e-4f);
    slo = max(slo, 0);
    shi = min(shi, NSTEP - 1);
    if (slo > shi) return false;

    // grid coords: f = (local+1)*0.5*(MD-1) = local*7.5 + 7.5
    pm.gb[0] = b0 * 7.5f + 7.5f; pm.gb[1] = b1 * 7.5f + 7.5f; pm.gb[2] = b2 * 7.5f + 7.5f;
    pm.gd[0] = d0 * 7.5f;        pm.gd[1] = d1 * 7.5f;        pm.gd[2] = d2 * 7.5f;
    pm.slo = slo; pm.shi = shi;
    return true;
}

__global__ __launch_bounds__(BLOCK)
void Raymarcher_kernel(const float* __restrict__ raypos,
                       const float* __restrict__ raydir,
                       const float* __restrict__ tminmax,
                       const float* __restrict__ primpos,
                       const float* __restrict__ primrot,
                       const float* __restrict__ primscale,
                       const float* __restrict__ tpl,
                       const float* __restrict__ dtp,
                       float* __restrict__ out)
{
    __shared__ __align__(16) float s_tpl[2][TPL_FLOATS]; // 128 KB, double-buffered
    __shared__ float4 s_scr[BLOCK * NSTEP];              // 96 KB per-thread step sums
    __shared__ float  s_pos[KK * 3];
    __shared__ float  s_rot[KK * 9];
    __shared__ float  s_scl[KK * 3];
    __shared__ float  s_rad2[KK];                        // bounding half-diagonal^2
    __shared__ unsigned s_mask[2];
    __shared__ int      s_qn;
    __shared__ unsigned char s_queue[KK];

    const int tid = threadIdx.x;
    const int bx = blockIdx.x % (WW / TW);
    const int by = blockIdx.x / (WW / TW);
    const int w = bx * TW + (tid % TW);
    const int h = by * THY + (tid / TW);
    const int ray = h * WW + w;

    const float px = raypos[ray * 3 + 0], py = raypos[ray * 3 + 1], pz = raypos[ray * 3 + 2];
    const float dx = raydir[ray * 3 + 0], dy = raydir[ray * 3 + 1], dz = raydir[ray * 3 + 2];
    const float tmn = tminmax[ray * 2 + 0], tmx = tminmax[ray * 2 + 1];
    const float dtv = dtp[0];

    // ---- init LDS: scratch, prim params, vote mask
    float4* sc = &s_scr[tid * NSTEP];
#pragma unroll
    for (int s = 0; s < NSTEP; ++s) sc[s] = make_float4(0.f, 0.f, 0.f, 0.f);
    for (int i = tid; i < KK * 3; i += BLOCK) s_pos[i] = primpos[i];
    for (int i = tid; i < KK * 9; i += BLOCK) s_rot[i] = primrot[i];
    for (int i = tid; i < KK * 3; i += BLOCK) s_scl[i] = primscale[i];
    if (tid < 2) s_mask[tid] = 0u;
    __syncthreads();
    if (tid < KK) {  // world-space bounding sphere: r^2 = 1/S0^2 + 1/S1^2 + 1/S2^2
        float i0 = 1.f / s_scl[tid * 3 + 0];
        float i1 = 1.f / s_scl[tid * 3 + 1];
        float i2 = 1.f / s_scl[tid * 3 + 2];
        s_rad2[tid] = i0 * i0 + i1 * i1 + i2 * i2;
    }
    __syncthreads();

    // ---- vote: which prims does any ray in this 16x8 tile intersect?
    // Per-thread hit bitmask avoids setup recompute later; lane0 batches the
    // workgroup-wide OR into 2 atomics per wave.
    unsigned myhit[2] = {0u, 0u};
    unsigned wvote[2] = {0u, 0u};
    for (int k = 0; k < KK; ++k) {
        // cheap conservative reject: squared line-to-center distance vs r^2
        float rx = s_pos[k * 3 + 0] - px;
        float ry = s_pos[k * 3 + 1] - py;
        float rz = s_pos[k * 3 + 2] - pz;
        float proj = rx * dx + ry * dy + rz * dz;
        float d2 = rx * rx + ry * ry + rz * rz - proj * proj;
        bool hit = false;
        if (__any(d2 <= s_rad2[k])) {
            PrimMarch pm;
            hit = (d2 <= s_rad2[k]) &&
                  prim_setup(k, s_pos, s_rot, s_scl, px, py, pz, dx, dy, dz,
                             tmn, tmx, dtv, pm);
        }
        if (hit) myhit[k >> 5] |= 1u << (k & 31);
        if (__any(hit)) wvote[k >> 5] |= 1u << (k & 31);
    }
    if ((tid & 31) == 0) {
        if (wvote[0]) atomicOr(&s_mask[0], wvote[0]);
        if (wvote[1]) atomicOr(&s_mask[1], wvote[1]);
    }
    __syncthreads();
    if (tid == 0) {
        int n = 0;
        for (int k = 0; k < KK; ++k)
            if ((s_mask[k >> 5] >> (k & 31)) & 1u) s_queue[n++] = (unsigned char)k;
        s_qn = n;
    }
    __syncthreads();
    const int qn = s_qn;

    // ---- double-buffered staging (TDM / async-LDS) + march
    if (qn > 0)
        stage(tpl + (size_t)s_queue[0] * TPL_FLOATS, s_tpl[0], tid);

    for (int qi = 0; qi < qn; ++qi) {
        stage_wait(tid);        // current buffer's DMA complete (issuing wave)
        __syncthreads();        // visible workgroup-wide
        if (qi + 1 < qn)        // overlap next prim's staging with this march
            stage(tpl + (size_t)s_queue[qi + 1] * TPL_FLOATS,
                  s_tpl[(qi + 1) & 1], tid);

        const int k = s_queue[qi];
        PrimMarch pm;
        if (((myhit[k >> 5] >> (k & 31)) & 1u) &&
            prim_setup(k, s_pos, s_rot, s_scl, px, py, pz, dx, dy, dz,
                       tmn, tmx, dtv, pm)) {
            const float* tb = s_tpl[qi & 1];
            float t0 = tmn + (float)pm.slo * dtv;
            float gx = pm.gb[0] + t0 * pm.gd[0];
            float gy = pm.gb[1] + t0 * pm.gd[1];
            float gz = pm.gb[2] + t0 * pm.gd[2];
            const float sx = pm.gd[0] * dtv, sy = pm.gd[1] * dtv, sz = pm.gd[2] * dtv;

            for (int s = pm.slo; s <= pm.shi; ++s) {
                float fx = fminf(fmaxf(gx, 0.f), 15.f);
                float fy = fminf(fmaxf(gy, 0.f), 15.f);
                float fz = fminf(fmaxf(gz, 0.f), 15.f);
                int ix = min((int)fx, MD - 2);
                int iy = min((int)fy, MD - 2);
                int iz = min((int)fz, MD - 2);
                float wx = fx - (float)ix, wy = fy - (float)iy, wz = fz - (float)iz;
                float wx0 = 1.f - wx, wy0 = 1.f - wy, wz0 = 1.f - wz;
                float w000 = wz0 * wy0 * wx0, w001 = wz0 * wy0 * wx;
                float w010 = wz0 * wy  * wx0, w011 = wz0 * wy  * wx;
                float w100 = wz  * wy0 * wx0, w101 = wz  * wy0 * wx;
                float w110 = wz  * wy  * wx0, w111 = wz  * wy  * wx;

                const float* p = tb + ((iz * MD + iy) * MD + ix);
                float4 a;
#pragma unroll
                for (int c = 0; c < 4; ++c) {
                    const float* q = p + c * (MD * MD * MD);
                    float v = q[0]   * w000 + q[1]   * w001
                            + q[16]  * w010 + q[17]  * w011
                            + q[256] * w100 + q[257] * w101
                            + q[272] * w110 + q[273] * w111;
                    ((float*)&a)[c] = v;
                }
                float4 o = sc[s];
                o.x += a.x; o.y += a.y; o.z += a.z; o.w += a.w;
                sc[s] = o;
                gx += sx; gy += sy; gz += sz;
            }
        }
        __syncthreads();        // everyone done reading buffer before next overwrite
    }

    // ---- sequential front-to-back composite (thread-private scratch)
    float r = 0.f, g = 0.f, b = 0.f, alpha = 0.f;
#pragma unroll
    for (int s = 0; s < NSTEP; ++s) {
        float4 v = sc[s];
        float contrib = fminf(alpha + v.w * dtv, 1.f) - alpha;
        r += v.x * contrib; g += v.y * contrib; b += v.z * contrib;
        alpha += contrib;
    }
    out[0 * HH * WW + ray] = r;
    out[1 * HH * WW + ray] = g;
    out[2 * HH * WW + ray] = b;
    out[3 * HH * WW + ray] = alpha;
}

extern "C" void kernel_launch(void* const* d_in, const int* in_sizes, int n_in,
                              void* d_out, int out_size, void* d_ws, size_t ws_size,
                              hipStream_t stream) {
    (void)in_sizes; (void)n_in; (void)d_ws; (void)ws_size; (void)out_size;
    const float* raypos    = (const float*)d_in[0];
    const float* raydir    = (const float*)d_in[1];
    const float* tminmax   = (const float*)d_in[2];
    const float* primpos   = (const float*)d_in[3];
    const float* primrot   = (const float*)d_in[4];
    const float* primscale = (const float*)d_in[5];
    const float* tpl       = (const float*)d_in[6];
    const float* dtp       = (const float*)d_in[7];
    float* out = (float*)d_out;

    dim3 grid((WW / TW) * (HH / THY));   // 128 blocks
    dim3 block(BLOCK);                   // 128 threads
    Raymarcher_kernel<<<grid, block, 0, stream>>>(
        raypos, raydir, tminmax, primpos, primrot, primscale, tpl, dtp, out);
}